// LCOT_61160334295353
// MI455X (gfx1250) — compile-verified
//
#include <hip/hip_runtime.h>
#include <math.h>

// LCOT circular-embedding distance, fused per-row in LDS.
// Row kernel: 1 block per (b,l) row pair -> sqrt(sum circ^2) into d_ws[row].
// Finalize kernel: deterministic fixed-order mean over L per batch -> d_out[b].

#define BB 16
#define LL 512
#define NS 512          // samples per row
#define NE 3000         // ECDF grid points (3*N_SAMPLES... actually 3*1000? no: 3*N_REF? ref uses 3*N where N=N_SAMPLES=512? -> 3*N with N = samples? See below)
#define NR 1000         // N_REF
#define TPB 256

// NOTE on NE: reference builds xnew = linspace(-1,2,3*N) with N = ref.shape[0] = N_REF?
// In emb(), N = ref.shape[0] = 1000 -> 3N = 3000. NE = 3000. NS = 512 samples.

typedef __attribute__((ext_vector_type(2))) float v2f;
typedef __attribute__((ext_vector_type(8))) float v8f;

__device__ __forceinline__ int lower_bound_lds(const float* a, int n, float q) {
  // first index i with a[i] >= q   (== jnp.searchsorted(a, q, side='left'))
  int lo = 0, hi = n;
  while (lo < hi) {
    int mid = (lo + hi) >> 1;
    if (a[mid] < q) lo = mid + 1;
    else hi = mid;
  }
  return lo;
}

__device__ __forceinline__ int clampi(int v, int lo, int hi) {
  return v < lo ? lo : (v > hi ? hi : v);
}

// interp1d of the ECDF (ys = (i+1)/NS) over sorted samples xs at query r,
// with linear extrapolation at both ends (torchinterp1d semantics).
__device__ __forceinline__ float ecdf_interp(const float* xs, float r) {
  int idx = lower_bound_lds(xs, NS, r);
  int i = clampi(idx - 1, 0, NS - 2);
  float x0 = xs[i];
  float w  = xs[i + 1] - x0;
  return (float)(i + 1) * (1.0f / NS) + (1.0f / NS) * (r - x0) / w;
}

// inverse-CDF: interp1d(ecdf_vals, xnew, q) where xnew[i] = -1 + i*step.
__device__ __forceinline__ float quantile_interp(const float* E, float q, float step) {
  int idx = lower_bound_lds(E, NE, q);
  int i = clampi(idx - 1, 0, NE - 2);
  float e0 = E[i];
  float y0 = -1.0f + step * (float)i;
  return y0 + step * (q - e0) / (E[i + 1] - e0);
}

__global__ __launch_bounds__(TPB)
void lcot_row_kernel(const float* __restrict__ x1,
                     const float* __restrict__ x2,
                     const float* __restrict__ ref,
                     float* __restrict__ row_out) {
  __shared__ float s1[NS];
  __shared__ float s2[NS];
  __shared__ float e1[NE];
  __shared__ float e2[NE];
  __shared__ float part[TPB];
  __shared__ float sh_a1, sh_a2;

  const unsigned t = threadIdx.x;
  const unsigned row = blockIdx.x;
  const float* g1 = x1 + (size_t)row * NS;
  const float* g2 = x2 + (size_t)row * NS;

  // ---- load both rows to LDS ----
  for (unsigned i = t; i < NS; i += TPB) {
    s1[i] = g1[i];
    s2[i] = g2[i];
  }
  __syncthreads();

  // ---- bitonic sort both rows in LDS (512 elements, 256 threads) ----
  for (unsigned k = 2; k <= NS; k <<= 1) {
    for (unsigned j = k >> 1; j > 0; j >>= 1) {
      unsigned i = ((t & ~(j - 1u)) << 1) | (t & (j - 1u));
      unsigned p = i | j;
      bool up = ((i & k) == 0);
      float a = s1[i], b = s1[p];
      if (up ? (a > b) : (a < b)) { s1[i] = b; s1[p] = a; }
      float c = s2[i], d = s2[p];
      if (up ? (c > d) : (c < d)) { s2[i] = d; s2[p] = c; }
      __syncthreads();
    }
  }

  // ---- alpha = mean(samples) - 0.5, fixed-order tree reduction ----
  part[t] = s1[t] + s1[t + TPB];
  __syncthreads();
  for (unsigned s = TPB / 2; s > 0; s >>= 1) {
    if (t < s) part[t] += part[t + s];
    __syncthreads();
  }
  if (t == 0) sh_a1 = part[0] * (1.0f / NS) - 0.5f;
  __syncthreads();
  part[t] = s2[t] + s2[t + TPB];
  __syncthreads();
  for (unsigned s = TPB / 2; s > 0; s >>= 1) {
    if (t < s) part[t] += part[t + s];
    __syncthreads();
  }
  if (t == 0) sh_a2 = part[0] * (1.0f / NS) - 0.5f;
  __syncthreads();

  // ---- circular ECDF on xnew = linspace(-1, 2, 3000) ----
  const float step = 3.0f / (float)(NE - 1);
  for (unsigned j = t; j < NE; j += TPB) {
    float xn = -1.0f + step * (float)j;
    float ix = floorf(xn);
    float rest = xn - ix;
    e1[j] = ix + ecdf_interp(s1, rest);
    e2[j] = ix + ecdf_interp(s2, rest);
  }
  __syncthreads();

  // ---- quantiles at ref - alpha, circular distance, per-thread partials ----
  const float a1 = sh_a1;
  const float a2 = sh_a2;
  float acc = 0.0f;
  for (unsigned kq = t; kq < NR; kq += TPB) {
    float rv = ref[kq];
    float v1 = quantile_interp(e1, rv - a1, step) - rv;   // emb1[k]
    float v2 = quantile_interp(e2, rv - a2, step) - rv;   // emb2[k]
    float d = fabsf(v2 - v1);
    float c = fminf(d, 1.0f - d);
    acc += c * c;
  }
  part[t] = acc;
  __syncthreads();

  // ---- reduce 256 partials; wave 0 uses V_WMMA_F32_16X16X4_F32 with B=ones:
  //      D[m][n] = sum_k A[m][k] + C[m][n]; 4 accumulating calls consume all 256.
  if (t < 32) {
#if __has_builtin(__builtin_amdgcn_wmma_f32_16x16x4_f32)
    v8f c = {};
    v2f b; b.x = 1.0f; b.y = 1.0f;
#pragma unroll
    for (int it = 0; it < 4; ++it) {
      v2f a;
      a.x = part[it * 64 + t];
      a.y = part[it * 64 + 32 + t];
      c = __builtin_amdgcn_wmma_f32_16x16x4_f32(
          /*neg_a=*/false, a, /*neg_b=*/false, b,
          /*c_mod=*/(short)0, c, /*reuse_a=*/false, /*reuse_b=*/false);
    }
    // every column of D holds the 16 row sums; per-lane sum of 8 VGPRs gives
    // M=0..7 (lanes 0-15) and M=8..15 (lanes 16-31) partial column sums.
    float s = c[0] + c[1] + c[2] + c[3] + c[4] + c[5] + c[6] + c[7];
    s += __shfl_xor(s, 16, 32);
#else
    // fallback: plain wave32 shuffle reduction
    float s = 0.0f;
#pragma unroll
    for (int it = 0; it < 8; ++it) s += part[it * 32 + t];
    for (int off = 16; off > 0; off >>= 1) s += __shfl_xor(s, off, 32);
#endif
    if (t == 0) row_out[row] = sqrtf(s);
  }
}

__global__ __launch_bounds__(TPB)
void lcot_finalize(const float* __restrict__ row_vals, float* __restrict__ out) {
  __shared__ float p[TPB];
  const unsigned b = blockIdx.x;
  const unsigned t = threadIdx.x;
  p[t] = row_vals[b * LL + t] + row_vals[b * LL + TPB + t];
  __syncthreads();
  for (unsigned s = TPB / 2; s > 0; s >>= 1) {
    if (t < s) p[t] += p[t + s];
    __syncthreads();
  }
  if (t == 0) out[b] = p[0] * (1.0f / (float)LL);
}

extern "C" void kernel_launch(void* const* d_in, const int* in_sizes, int n_in,
                              void* d_out, int out_size, void* d_ws, size_t ws_size,
                              hipStream_t stream) {
  const float* x1  = (const float*)d_in[0];
  const float* x2  = (const float*)d_in[1];
  const float* ref = (const float*)d_in[2];
  float* row_vals  = (float*)d_ws;      // B*L floats = 32 KB of scratch
  float* out       = (float*)d_out;     // 16 floats

  lcot_row_kernel<<<BB * LL, TPB, 0, stream>>>(x1, x2, ref, row_vals);
  lcot_finalize<<<BB, TPB, 0, stream>>>(row_vals, out);
}